// SampleK_14482629722272
// MI455X (gfx1250) — compile-verified
//
#include <hip/hip_runtime.h>

typedef __attribute__((ext_vector_type(2))) float v2f;
typedef __attribute__((ext_vector_type(8))) float v8f;

#define N2 8192
#define N1 4096
#define KNN 16
#define BIGF 3.402823466e38f

__global__ __launch_bounds__(256) void knn_wmma_kernel(const float* __restrict__ xyz2,
                                                       const float* __restrict__ xyz1,
                                                       int* __restrict__ out) {
  __shared__ float s_xx[N2];          // ||x2||^2 for this batch (32 KB)
  __shared__ float s_md[256 * KNN];   // per-lane candidate distances (16 KB)
  __shared__ int   s_mi[256 * KNN];   // per-lane candidate indices   (16 KB)

  const int tid  = threadIdx.x;
  const int lane = tid & 31;
  const int wv   = tid >> 5;
  const int half = lane >> 4;         // 0: lanes 0-15, 1: lanes 16-31
  const int l15  = lane & 15;

  const int bid   = blockIdx.x;
  const int b     = bid >> 5;         // 32 blocks per batch (128 queries each)
  const int qbase = (bid & 31) * 128;
  const int q0    = qbase + wv * 16;  // this wave's 16-query tile

  const float* x2b = xyz2 + (size_t)b * 3 * N2;
  const float* x1b = xyz1 + (size_t)b * 3 * N1;

  // Cooperative: ||x2||^2 into LDS
  for (int i = tid; i < N2; i += 256) {
    float a0 = x2b[i], a1 = x2b[N2 + i], a2 = x2b[2 * N2 + i];
    s_xx[i] = a0 * a0 + a1 * a1 + a2 * a2;
  }
  __syncthreads();

  // B operand (4x16 query tile, K padded 3->4): loop-invariant in 2 VGPRs
  const int n = q0 + l15;
  float y0 = x1b[n], y1 = x1b[N1 + n], y2 = x1b[2 * N1 + n];
  const float yy = y0 * y0 + y1 * y1 + y2 * y2;
  v2f bop;
  bop.x = half ? y2 : y0;
  bop.y = half ? 0.0f : y1;

  // Per-lane running top-16 (registers, fully unrolled access)
  float bd[KNN]; int bi_[KNN];
#pragma unroll
  for (int j = 0; j < KNN; ++j) { bd[j] = BIGF; bi_[j] = -1; }
  float curmax = BIGF;

  const float4* xx4 = (const float4*)s_xx;

  for (int tb = 0; tb < N2; tb += 16) {
    // A operand (16x4 source tile)
    const int m = tb + l15;
    v2f aop;
    aop.x = x2b[(half ? 2 * N2 : 0) + m];
    float ay = x2b[N2 + m];
    aop.y = half ? 0.0f : ay;

    // cross terms: D = A x B  (16x16 dot products)
    v8f c = {};
    c = __builtin_amdgcn_wmma_f32_16x16x4_f32(false, aop, false, bop,
                                              (short)0, c, false, false);

    // rows this lane holds: rbase..rbase+7
    const int rbase = tb + half * 8;
    float4 xa = xx4[rbase >> 2];
    float4 xb = xx4[(rbase >> 2) + 1];
    float xr[8] = {xa.x, xa.y, xa.z, xa.w, xb.x, xb.y, xb.z, xb.w};

#pragma unroll
    for (int r = 0; r < 8; ++r) {
      float d = fmaxf(xr[r] + yy - 2.0f * c[r], 0.0f);
      if (d < curmax) {
        // replace current max, then refresh curmax (all static-indexed)
        int pos = 0; float mx = bd[0];
#pragma unroll
        for (int j = 1; j < KNN; ++j) if (bd[j] > mx) { mx = bd[j]; pos = j; }
#pragma unroll
        for (int j = 0; j < KNN; ++j) if (j == pos) { bd[j] = d; bi_[j] = rbase + r; }
        mx = bd[0];
#pragma unroll
        for (int j = 1; j < KNN; ++j) mx = fmaxf(mx, bd[j]);
        curmax = mx;
      }
    }
  }

  // Dump per-lane top-16; lanes q and q+16 partition the candidate rows
#pragma unroll
  for (int j = 0; j < KNN; ++j) {
    s_md[tid * KNN + j] = bd[j];
    s_mi[tid * KNN + j] = bi_[j];
  }
  __syncthreads();

  if (half == 0) {
    float cd[2 * KNN]; int ci[2 * KNN];
    const int t0 = wv * 32 + l15;
    const int t1 = t0 + 16;
#pragma unroll
    for (int j = 0; j < KNN; ++j) { cd[j]       = s_md[t0 * KNN + j]; ci[j]       = s_mi[t0 * KNN + j]; }
#pragma unroll
    for (int j = 0; j < KNN; ++j) { cd[KNN + j] = s_md[t1 * KNN + j]; ci[KNN + j] = s_mi[t1 * KNN + j]; }

    int* orow = out + ((size_t)b * N1 + n) * KNN;
    for (int k = 0; k < KNN; ++k) {
      float mind = cd[0]; int mini = ci[0];
#pragma unroll
      for (int j = 1; j < 2 * KNN; ++j) {
        bool better = (cd[j] < mind) || ((cd[j] == mind) && (ci[j] < mini));
        if (better) { mind = cd[j]; mini = ci[j]; }
      }
      orow[k] = mini;
#pragma unroll
      for (int j = 0; j < 2 * KNN; ++j) if (ci[j] == mini) cd[j] = BIGF;
    }
  }
}

extern "C" void kernel_launch(void* const* d_in, const int* in_sizes, int n_in,
                              void* d_out, int out_size, void* d_ws, size_t ws_size,
                              hipStream_t stream) {
  const float* xyz2 = (const float*)d_in[0];  // [8,3,8192]
  const float* xyz1 = (const float*)d_in[1];  // [8,3,4096]
  int* out = (int*)d_out;                     // [8,4096,16] int32
  (void)in_sizes; (void)n_in; (void)out_size; (void)d_ws; (void)ws_size;
  knn_wmma_kernel<<<dim3(256), dim3(256), 0, stream>>>(xyz2, xyz1, out);
}